// SparseMultiHeadAttention_65850438582298
// MI455X (gfx1250) — compile-verified
//
#include <hip/hip_runtime.h>

// ---------------------------------------------------------------------------
// CDNA5 (gfx1250) implementation of SparseMultiHeadAttention reference.
// - All GEMM-shaped math uses V_WMMA_F32_16X16X4_F32 (fp32, wave32).
// - LDS tile staging uses GLOBAL_LOAD_ASYNC_TO_LDS_B128 (ASYNCcnt path).
// - Streaming kernels prefetch the next tile (global_prefetch_b8).
// ---------------------------------------------------------------------------

typedef __attribute__((ext_vector_type(2))) float v2f;
typedef __attribute__((ext_vector_type(8))) float v8f;

#define WMMA_F32(a, b, c) \
  __builtin_amdgcn_wmma_f32_16x16x4_f32(false, (a), false, (b), (short)0, (c), false, false)

// ---------------- async global->LDS copy (CDNA5) ----------------
// LDS byte address = low 32 bits of the generic pointer to a __shared__ object
// (same addressing DS instructions use). 16 bytes per lane per issue.
__device__ __forceinline__ void async_copy_b128(unsigned lds_addr, const void* gptr) {
  asm volatile("global_load_async_to_lds_b128 %0, %1, off"
               :: "v"(lds_addr), "v"((unsigned long long)(uintptr_t)gptr)
               : "memory");
}
__device__ __forceinline__ void async_wait0() {
  asm volatile("s_wait_asynccnt 0x0" ::: "memory");
}
__device__ __forceinline__ unsigned lds_addr_of(const void* p) {
  return (unsigned)(uintptr_t)p;
}

// ---------------- wave helpers (wave32) ----------------
__device__ __forceinline__ float wave_sum32(float v) {
  for (int o = 16; o > 0; o >>= 1) v += __shfl_xor(v, o, 32);
  return v;
}
// reductions within a 16-lane half (bit4 untouched by offsets <16)
__device__ __forceinline__ float half_max16(float v) {
  for (int o = 8; o > 0; o >>= 1) v = fmaxf(v, __shfl_xor(v, o, 32));
  return v;
}
__device__ __forceinline__ float half_sum16(float v) {
  for (int o = 8; o > 0; o >>= 1) v += __shfl_xor(v, o, 32);
  return v;
}
__device__ __forceinline__ void half_argmax16(float& v, int& idx) {
  for (int o = 8; o > 0; o >>= 1) {
    float ov = __shfl_xor(v, o, 32);
    int oi = __shfl_xor(idx, o, 32);
    if (ov > v || (ov == v && oi < idx)) { v = ov; idx = oi; }
  }
}

// ---------------- tiny init ----------------
__global__ void init_scalars_k(float* scal) {
  if (threadIdx.x < 4) scal[threadIdx.x] = 0.0f;
}

// ---------------- generic fp32 WMMA GEMM:  C = A[MxK] * B[KxN] + bias ------
// block = 128 threads (4 waves); each wave computes a 16x64 strip.
__global__ void gemm_bias_wmma(const float* __restrict__ A, const float* __restrict__ B,
                               const float* __restrict__ bias, float* __restrict__ Cm,
                               int M, int Nn, int K) {
  const int lane = threadIdx.x & 31;
  const int wave = threadIdx.x >> 5;
  const int ntiles = Nn >> 6;           // 64-wide strips
  const int gw = blockIdx.x * 4 + wave;
  const int mbase = (gw / ntiles) * 16;
  const int nbase = (gw % ntiles) * 64;
  const int laneM = lane & 15;
  const int hi = lane >> 4;

  v8f acc0 = {}, acc1 = {}, acc2 = {}, acc3 = {};
  const float* arow = A + (size_t)(mbase + laneM) * K;
  for (int k0 = 0; k0 < K; k0 += 4) {
    v2f a;
    a.x = arow[k0 + 2 * hi];
    a.y = arow[k0 + 2 * hi + 1];
    const float* b0 = B + (size_t)(k0 + 2 * hi) * Nn + nbase + laneM;
    const float* b1 = B + (size_t)(k0 + 2 * hi + 1) * Nn + nbase + laneM;
    // cover latency of the strided B stream: prefetch 32 K-steps ahead
    if ((k0 & 127) == 0 && k0 + 128 < K)
      __builtin_prefetch(B + (size_t)(k0 + 128 + 2 * hi) * Nn + nbase + laneM, 0, 0);
    v2f b;
    b.x = b0[0];  b.y = b1[0];  acc0 = WMMA_F32(a, b, acc0);
    b.x = b0[16]; b.y = b1[16]; acc1 = WMMA_F32(a, b, acc1);
    b.x = b0[32]; b.y = b1[32]; acc2 = WMMA_F32(a, b, acc2);
    b.x = b0[48]; b.y = b1[48]; acc3 = WMMA_F32(a, b, acc3);
  }
  const int row0 = mbase + hi * 8;
#pragma unroll
  for (int r = 0; r < 8; r++) {
    size_t o = (size_t)(row0 + r) * Nn + nbase + laneM;
    Cm[o]      = acc0[r] + bias[nbase + laneM];
    Cm[o + 16] = acc1[r] + bias[nbase + 16 + laneM];
    Cm[o + 32] = acc2[r] + bias[nbase + 32 + laneM];
    Cm[o + 48] = acc3[r] + bias[nbase + 48 + laneM];
  }
}

// ---------------- split qkv, apply mh_rms(gamma_q) to q --------------------
// one wave per (n,h) segment of C=64 elems; block = 128 (4 waves)
__global__ void split_qkv_rms(const float* __restrict__ qkv, const float* __restrict__ gq,
                              float* __restrict__ q, float* __restrict__ kf,
                              float* __restrict__ vf) {
  const int lane = threadIdx.x & 31;
  const int seg = blockIdx.x * 4 + (threadIdx.x >> 5);  // 0..N*H-1
  const int n = seg >> 4, h = seg & 15;
  const float* base = qkv + (size_t)n * 3072 + h * 64;
  const float q0 = base[lane], q1 = base[lane + 32];
  const float ka = base[1024 + lane], kb = base[1024 + lane + 32];
  const float va = base[2048 + lane], vb = base[2048 + lane + 32];
  const float ss = wave_sum32(q0 * q0 + q1 * q1);
  const float inv = 8.0f / fmaxf(sqrtf(ss), 1e-12f);  // sqrt(C)=8
  const size_t o = (size_t)n * 1024 + h * 64;
  q[o + lane]      = q0 * inv * gq[h * 64 + lane];
  q[o + lane + 32] = q1 * inv * gq[h * 64 + lane + 32];
  kf[o + lane] = ka; kf[o + lane + 32] = kb;
  vf[o + lane] = va; vf[o + lane + 32] = vb;
}

// ---------------- per-row l2 normalization (rows of 1024) ------------------
__global__ void l2norm_rows(const float* __restrict__ in, float* __restrict__ out) {
  __shared__ float red[256];
  const int t = threadIdx.x;
  const float* r = in + (size_t)blockIdx.x * 1024;
  float s = 0.0f;
#pragma unroll
  for (int i = 0; i < 4; i++) { float v = r[t + i * 256]; s += v * v; }
  red[t] = s; __syncthreads();
  for (int o = 128; o > 0; o >>= 1) { if (t < o) red[t] += red[t + o]; __syncthreads(); }
  const float inv = 1.0f / fmaxf(sqrtf(red[0]), 1e-12f);
  float* w = out + (size_t)blockIdx.x * 1024;
#pragma unroll
  for (int i = 0; i < 4; i++) w[t + i * 256] = r[t + i * 256] * inv;
}

// ---------------- sim GEMM + fused per-row argmax --------------------------
// one wave per 16 rows; A-strip (16x1024 = 64 KB) staged into LDS with
// async global->LDS B128 copies; stream 256 column tiles with WMMA,
// keep running (max, argmax) with first-index tie-breaking.
__global__ void sim_argmax(const float* __restrict__ kbn, const float* __restrict__ side,
                           int* __restrict__ idx_out, float* __restrict__ val_out) {
  __shared__ float sA[16 * 1024];
  const int lane = threadIdx.x;
  const int mbase = blockIdx.x * 16;
  const char* src = (const char*)(kbn + (size_t)mbase * 1024);  // 16 contiguous rows
  const unsigned lbase = lds_addr_of(sA);
  for (int i = lane; i < 4096; i += 32)           // 4096 x 16B = 64 KB
    async_copy_b128(lbase + i * 16, src + (size_t)i * 16);
  async_wait0();
  __syncthreads();

  const int laneM = lane & 15;
  const int hi = lane >> 4;
  float best[8]; int bidx[8];
#pragma unroll
  for (int r = 0; r < 8; r++) { best[r] = -3.0e38f; bidx[r] = 0; }

  const float* arow = sA + laneM * 1024;
  for (int ct = 0; ct < 256; ct++) {
    const int cb = ct * 16;
    const float* bcol = side + (size_t)(cb + laneM) * 1024;  // B = side^T
    if (ct + 1 < 256)                       // pull next tile's rows toward L2/L0
      __builtin_prefetch(side + (size_t)(cb + 16 + laneM) * 1024, 0, 0);
    v8f acc = {};
    for (int k0 = 0; k0 < 1024; k0 += 4) {
      v2f a; a.x = arow[k0 + 2 * hi]; a.y = arow[k0 + 2 * hi + 1];
      v2f b; b.x = bcol[k0 + 2 * hi]; b.y = bcol[k0 + 2 * hi + 1];
      acc = WMMA_F32(a, b, acc);
    }
    const int col = cb + laneM;
#pragma unroll
    for (int r = 0; r < 8; r++) {
      if (acc[r] > best[r]) { best[r] = acc[r]; bidx[r] = col; }  // keeps first (cols ascend)
    }
  }
#pragma unroll
  for (int r = 0; r < 8; r++) {
    float v = best[r]; int i = bidx[r];
    half_argmax16(v, i);
    if (laneM == 0) {
      const int row = mbase + hi * 8 + r;
      idx_out[row] = i;
      val_out[row] = v;
    }
  }
}

// ---------------- gather + combine + Frobenius-norm accumulation -----------
__global__ void combine_norms(const float* __restrict__ kf, const float* __restrict__ vf,
                              const float* __restrict__ k0f, const float* __restrict__ v0f,
                              const float* __restrict__ k1f, const float* __restrict__ v1f,
                              const int* __restrict__ idx0, const float* __restrict__ val0,
                              const int* __restrict__ idx1, const float* __restrict__ val1,
                              float* __restrict__ ck, float* __restrict__ cv,
                              float* __restrict__ scal) {
  __shared__ float red[4][256];
  const int t = threadIdx.x;
  const size_t e = (size_t)blockIdx.x * 256 + t;
  const int n = (int)(e >> 10);
  const int d = (int)(e & 1023);
  const float kfe = kf[e], vfe = vf[e];
  const bool va = val0[n] > 0.6f; const int i0 = idx0[n];
  const bool vb = val1[n] > 0.6f; const int i1 = idx1[n];
  const float ka  = va ? k0f[(size_t)i0 * 1024 + d] : kfe;
  const float v0e = va ? v0f[(size_t)i0 * 1024 + d] : vfe;
  const float kb  = vb ? k1f[(size_t)i1 * 1024 + d] : kfe;
  const float v1e = vb ? v1f[(size_t)i1 * 1024 + d] : vfe;
  const float ckv = 0.5f * ka + 0.5f * kb + kfe;   // (1-ALPHA)=ALPHA=0.5
  const float cvv = 0.5f * v0e + 0.5f * v1e + vfe;
  ck[e] = ckv; cv[e] = cvv;
  red[0][t] = kfe * kfe; red[1][t] = ckv * ckv;
  red[2][t] = vfe * vfe; red[3][t] = cvv * cvv;
  __syncthreads();
  for (int o = 128; o > 0; o >>= 1) {
    if (t < o) {
      red[0][t] += red[0][t + o]; red[1][t] += red[1][t + o];
      red[2][t] += red[2][t + o]; red[3][t] += red[3][t + o];
    }
    __syncthreads();
  }
  if (t == 0) {
    atomicAdd(&scal[0], red[0][0]); atomicAdd(&scal[1], red[1][0]);
    atomicAdd(&scal[2], red[2][0]); atomicAdd(&scal[3], red[3][0]);
  }
}

// ---------------- apply global scale; mh_rms(gamma_k) on k; scale v --------
__global__ void scale_rms(float* __restrict__ ck, float* __restrict__ cv,
                          const float* __restrict__ gk, const float* __restrict__ scal) {
  const int lane = threadIdx.x & 31;
  const int seg = blockIdx.x * 4 + (threadIdx.x >> 5);
  const int n = seg >> 4, h = seg & 15;
  const float sk = sqrtf(scal[0] / fmaxf(scal[1], 1e-30f));
  const float sv = sqrtf(scal[2] / fmaxf(scal[3], 1e-30f));
  const size_t o = (size_t)n * 1024 + h * 64;
  const float a = ck[o + lane] * sk, b = ck[o + lane + 32] * sk;
  const float ss = wave_sum32(a * a + b * b);
  const float inv = 8.0f / fmaxf(sqrtf(ss), 1e-12f);
  ck[o + lane]      = a * inv * gk[h * 64 + lane];
  ck[o + lane + 32] = b * inv * gk[h * 64 + lane + 32];
  cv[o + lane]      *= sv;
  cv[o + lane + 32] *= sv;
}

// ---------------- flash attention per head (online softmax) ----------------
// grid = H * (N/16) blocks, 1 wave each. Q tile staged to LDS with async
// B128 copies; P tile round-trips through LDS (D-layout -> A-layout).
__global__ void attention_flash(const float* __restrict__ q, const float* __restrict__ kk,
                                const float* __restrict__ vv, float* __restrict__ hout) {
  __shared__ float sQ[16 * 64];
  __shared__ float sP[16][18];
  const int lane = threadIdx.x;
  const int h = blockIdx.x >> 8;
  const int mbase = (blockIdx.x & 255) * 16;
  {
    const unsigned lq = lds_addr_of(sQ);
    // 16 rows x 64 floats; each row segment = 256 B = 16 x B128
    for (int i = lane; i < 256; i += 32) {
      const int r = i >> 4, c16 = i & 15;
      const char* g = (const char*)(q + (size_t)(mbase + r) * 1024 + h * 64 + c16 * 4);
      async_copy_b128(lq + i * 16, g);
    }
    async_wait0();
  }
  __syncthreads();

  const int laneM = lane & 15;
  const int hi = lane >> 4;
  v8f o0 = {}, o1 = {}, o2 = {}, o3 = {};
  float mrow[8], lrow[8];
#pragma unroll
  for (int r = 0; r < 8; r++) { mrow[r] = -3.0e38f; lrow[r] = 0.0f; }

  const float* aq = sQ + laneM * 64;
  for (int kt = 0; kt < 256; kt++) {
    const int kb = kt * 16;
    if (kt + 1 < 256) {   // warm next K/V rows
      __builtin_prefetch(kk + (size_t)(kb + 16 + laneM) * 1024 + h * 64, 0, 0);
      __builtin_prefetch(vv + (size_t)(kb + 16 + laneM) * 1024 + h * 64, 0, 0);
    }
    // scores tile S = Q (16x64) * K^T (64x16)
    v8f s = {};
    const float* bk = kk + (size_t)(kb + laneM) * 1024 + h * 64;
#pragma unroll
    for (int k0 = 0; k0 < 64; k0 += 4) {
      v2f a; a.x = aq[k0 + 2 * hi]; a.y = aq[k0 + 2 * hi + 1];
      v2f b; b.x = bk[k0 + 2 * hi]; b.y = bk[k0 + 2 * hi + 1];
      s = WMMA_F32(a, b, s);
    }
    // online softmax update
    float corr[8];
#pragma unroll
    for (int r = 0; r < 8; r++) {
      const float sv = s[r] * 0.125f;                 // 1/sqrt(C)
      const float tm = half_max16(sv);
      const float mn = fmaxf(mrow[r], tm);
      corr[r] = __expf(mrow[r] - mn);
      const float p = __expf(sv - mn);
      lrow[r] = lrow[r] * corr[r] + half_sum16(p);
      mrow[r] = mn;
      sP[hi * 8 + r][laneM] = p;                      // D-layout -> LDS
    }
    __syncthreads();
#pragma unroll
    for (int r = 0; r < 8; r++) { o0[r] *= corr[r]; o1[r] *= corr[r]; o2[r] *= corr[r]; o3[r] *= corr[r]; }
    // O += P (16x16) * V (16x64)
#pragma unroll
    for (int k0 = 0; k0 < 16; k0 += 4) {
      v2f a; a.x = sP[laneM][k0 + 2 * hi]; a.y = sP[laneM][k0 + 2 * hi + 1];
      const float* bv0 = vv + (size_t)(kb + k0 + 2 * hi) * 1024 + h * 64 + laneM;
      const float* bv1 = vv + (size_t)(kb + k0 + 2 * hi + 1) * 1024 + h * 64 + laneM;
      v2f b;
      b.x = bv0[0];  b.y = bv1[0];  o0 = WMMA_F32(a, b, o0);
      b.x = bv0[16]; b.y = bv1[16]; o1 = WMMA_F32(a, b, o1);
      b.x = bv0[32]; b.y = bv1[32]; o2 = WMMA_F32(a, b, o2);
      b.x = bv0[48]; b.y = bv1[48]; o3 = WMMA_F32(a, b, o3);
    }
    __syncthreads();
  }
#pragma unroll
  for (int r = 0; r < 8; r++) {
    const float inv = 1.0f / lrow[r];
    const size_t o = (size_t)(mbase + hi * 8 + r) * 1024 + h * 64 + laneM;
    hout[o]      = o0[r] * inv;
    hout[o + 16] = o1[r] * inv;
    hout[o + 32] = o2[r] * inv;
    hout[o + 48] = o3[r] * inv;
  }
}

// ---------------------------------------------------------------------------
extern "C" void kernel_launch(void* const* d_in, const int* in_sizes, int n_in,
                              void* d_out, int out_size, void* d_ws, size_t ws_size,
                              hipStream_t stream) {
  (void)in_sizes; (void)n_in; (void)out_size; (void)ws_size;
  const float* x    = (const float*)d_in[0];
  const float* k0   = (const float*)d_in[1];
  const float* v0   = (const float*)d_in[2];
  const float* k1   = (const float*)d_in[3];
  const float* v1   = (const float*)d_in[4];
  const float* Wqkv = (const float*)d_in[5];
  const float* bqkv = (const float*)d_in[6];
  const float* gq   = (const float*)d_in[7];
  const float* gk   = (const float*)d_in[8];
  const float* Wout = (const float*)d_in[9];
  const float* bout = (const float*)d_in[10];

  const size_t ND = (size_t)4096 * 1024;
  float* ws  = (float*)d_ws;
  float* qkv = ws;               // 3*ND
  float* q   = qkv + 3 * ND;     // ND
  float* kf  = q + ND;           // ND
  float* vf  = kf + ND;          // ND
  float* kbn = vf + ND;          // ND
  float* k0n = kbn + ND;         // ND
  float* k1n = k0n + ND;         // ND
  float* ck  = k1n + ND;         // ND (becomes k_out after scale_rms)
  float* cv  = ck + ND;          // ND (becomes v_out)
  float* hb  = cv + ND;          // ND
  float* val0 = hb + ND;         // 4096
  float* val1 = val0 + 4096;     // 4096
  float* scal = val1 + 4096;     // 4 (+pad)
  int* idx0 = (int*)(scal + 16); // 4096
  int* idx1 = idx0 + 4096;       // 4096

  init_scalars_k<<<1, 32, 0, stream>>>(scal);

  // qkv = x @ W_qkv + b_qkv   (M=4096, N=3072, K=1024)
  gemm_bias_wmma<<<3072, 128, 0, stream>>>(x, Wqkv, bqkv, qkv, 4096, 3072, 1024);

  split_qkv_rms<<<16384, 128, 0, stream>>>(qkv, gq, q, kf, vf);

  l2norm_rows<<<4096, 256, 0, stream>>>(kf, kbn);
  l2norm_rows<<<4096, 256, 0, stream>>>(k0, k0n);
  l2norm_rows<<<4096, 256, 0, stream>>>(k1, k1n);

  sim_argmax<<<256, 32, 0, stream>>>(kbn, k0n, idx0, val0);
  sim_argmax<<<256, 32, 0, stream>>>(kbn, k1n, idx1, val1);

  combine_norms<<<16384, 256, 0, stream>>>(kf, vf, k0, v0, k1, v1,
                                           idx0, val0, idx1, val1, ck, cv, scal);
  scale_rms<<<16384, 128, 0, stream>>>(ck, cv, gk, scal);

  attention_flash<<<4096, 32, 0, stream>>>(q, ck, cv, hb);

  // out = h @ W_out + b_out   (M=4096, N=1024, K=1024)
  gemm_bias_wmma<<<1024, 128, 0, stream>>>(hb, Wout, bout, (float*)d_out, 4096, 1024, 1024);
}